// MultiHeadAttention_5351529251164
// MI455X (gfx1250) — compile-verified
//
#include <hip/hip_runtime.h>
#include <hip/hip_bf16.h>
#include <math.h>

// Problem constants (reference: B,S,D,H = 4,2048,768,12; HD=64)
constexpr int kB = 4;
constexpr int kS = 2048;
constexpr int kD = 768;
constexpr int kH = 12;
constexpr int kHD = 64;

typedef __attribute__((ext_vector_type(2))) float v2f;
typedef __attribute__((ext_vector_type(8))) float v8f;

// D = A(16x4) * B(4x16) + C, fp32 WMMA (CDNA5: V_WMMA_F32_16X16X4_F32)
__device__ __forceinline__ v8f wmma4(v2f a, v2f b, v8f c) {
  return __builtin_amdgcn_wmma_f32_16x16x4_f32(
      /*neg_a=*/false, a, /*neg_b=*/false, b,
      /*c_mod=*/(short)0, c, /*reuse_a=*/false, /*reuse_b=*/false);
}

// ---------------------------------------------------------------------------
// Tiled GEMM: Y[M,N] = A[M,K] @ W[K,N] (+ bias[N]). 256 threads = 8 waves.
// 64x64 output tile per workgroup; each wave computes two 16x16 C tiles.
// ---------------------------------------------------------------------------
#define GTM 64
#define GTN 64
#define GKB 16
#define GLDA 20   // 16 + 4 pad  (row stride, floats; 80B -> float4 aligned)
#define GLDB 68   // 64 + 4 pad  (272B -> float4 aligned)

__global__ __launch_bounds__(256)
void gemm_bias_kernel(const float* __restrict__ A, const float* __restrict__ W,
                      const float* __restrict__ bias, float* __restrict__ Y,
                      int M, int N, int K) {
  __shared__ float As[GTM * GLDA];
  __shared__ float Bs[GKB * GLDB];

  const int tid   = threadIdx.x;
  const int lane  = tid & 31;
  const int wave  = tid >> 5;          // 0..7
  const int lnlo  = lane & 15;
  const int khalf = lane >> 4;         // 0: lanes 0-15, 1: lanes 16-31

  const int row0 = blockIdx.y * GTM;
  const int col0 = blockIdx.x * GTN;
  const int wrow = (wave & 3) * 16;    // row strip within tile
  const int wcol = (wave >> 2) * 32;   // two 16-col tiles: wcol, wcol+16

  v8f c0 = {}; v8f c1 = {};

  for (int k0 = 0; k0 < K; k0 += GKB) {
    __syncthreads();
    {   // stage A tile 64 x 16 (coalesced float4)
      const int r  = tid >> 2;         // 0..63
      const int c4 = tid & 3;          // 0..3
      const float4 av = *(const float4*)(A + (size_t)(row0 + r) * K + k0 + c4 * 4);
      *(float4*)(&As[r * GLDA + c4 * 4]) = av;
    }
    {   // stage W tile 16 x 64
      const int r  = tid >> 4;         // 0..15
      const int c4 = tid & 15;         // 0..15
      const float4 wv = *(const float4*)(W + (size_t)(k0 + r) * N + col0 + c4 * 4);
      *(float4*)(&Bs[r * GLDB + c4 * 4]) = wv;
    }
    if (k0 + GKB < K) {  // hint next A chunk into cache (global_prefetch_b8)
      __builtin_prefetch(A + (size_t)(row0 + (tid >> 2)) * K + k0 + GKB, 0, 1);
    }
    __syncthreads();

#pragma unroll
    for (int kk = 0; kk < GKB; kk += 4) {
      const int kA = kk + khalf * 2;   // A/B fragment K index per ISA layout
      v2f a, b0, b1;
      a.x  = As[(wrow + lnlo) * GLDA + kA];
      a.y  = As[(wrow + lnlo) * GLDA + kA + 1];
      b0.x = Bs[(kA + 0) * GLDB + wcol + lnlo];
      b0.y = Bs[(kA + 1) * GLDB + wcol + lnlo];
      b1.x = Bs[(kA + 0) * GLDB + wcol + 16 + lnlo];
      b1.y = Bs[(kA + 1) * GLDB + wcol + 16 + lnlo];
      c0 = wmma4(a, b0, c0);
      c1 = wmma4(a, b1, c1);
    }
  }

  const float bv0 = bias ? bias[col0 + wcol + lnlo]      : 0.f;
  const float bv1 = bias ? bias[col0 + wcol + 16 + lnlo] : 0.f;
  const int rbase = row0 + wrow + khalf * 8;   // C layout: upper lanes = M+8
#pragma unroll
  for (int r = 0; r < 8; ++r) {
    Y[(size_t)(rbase + r) * N + col0 + wcol + lnlo]      = c0[r] + bv0;
    Y[(size_t)(rbase + r) * N + col0 + wcol + 16 + lnlo] = c1[r] + bv1;
  }
}

// ---------------------------------------------------------------------------
// Causal flash attention. Grid: (S/64, H, B). Block: 128 threads = 4 waves.
// Each wave owns a 16-row q strip; workgroup handles 64 q rows of one head.
// Q,K,V,Ctx are [B,S,D] fp32; head h occupies columns [h*64, h*64+64).
// ---------------------------------------------------------------------------
#define ALDK 66   // pad for P strip
#define ALDV 68   // pad for K/V tiles (272B, float4 aligned)

__global__ __launch_bounds__(128)
void flash_attn_kernel(const float* __restrict__ Q, const float* __restrict__ Kg,
                       const float* __restrict__ Vg, float* __restrict__ Ctx) {
  __shared__ float Ks[64 * ALDV];
  __shared__ float Vs[64 * ALDV];
  __shared__ float Ps[4 * 16 * ALDK];   // wave-private P strips

  const int tid   = threadIdx.x;
  const int lane  = tid & 31;
  const int wave  = tid >> 5;          // 0..3
  const int lnlo  = lane & 15;
  const int khalf = lane >> 4;

  const int qt = blockIdx.x;           // q tile (64 rows)
  const int h  = blockIdx.y;
  const int b  = blockIdx.z;
  const size_t bo = (size_t)b * kS * kD;
  const int hc = h * kHD;
  const int qbase = qt * 64 + wave * 16;

  // Q strip (16 x 64) straight into A-fragment registers: 16 K-steps of v2f.
  v2f aq[16];
  {
    const float* qp = Q + bo + (size_t)(qbase + lnlo) * kD + hc;
#pragma unroll
    for (int kk = 0; kk < 16; ++kk) {
      const int d = kk * 4 + khalf * 2;
      aq[kk].x = qp[d];
      aq[kk].y = qp[d + 1];
    }
  }

  v8f o0 = {}, o1 = {}, o2 = {}, o3 = {};
  float mrow[8], lrow[8];
#pragma unroll
  for (int r = 0; r < 8; ++r) { mrow[r] = -__builtin_inff(); lrow[r] = 0.f; }

  float* Pw = &Ps[wave * 16 * ALDK];

  for (int kt = 0; kt <= qt; ++kt) {
    const int kbase = kt * 64;
    __syncthreads();                    // protect K/V from previous iteration
#pragma unroll
    for (int i = 0; i < 8; ++i) {       // stage K,V 64x64 head slices (float4)
      const int e  = tid + i * 128;     // float4 element 0..1023
      const int r  = e >> 4;
      const int c4 = e & 15;
      const size_t g = bo + (size_t)(kbase + r) * kD + hc + c4 * 4;
      const float4 kv = *(const float4*)(Kg + g);
      const float4 vv = *(const float4*)(Vg + g);
      *(float4*)(&Ks[r * ALDV + c4 * 4]) = kv;
      *(float4*)(&Vs[r * ALDV + c4 * 4]) = vv;
    }
    __syncthreads();

    // ---- scores S = Q K^T : four 16x16 tiles, K-dim = 64 (16 WMMAs each)
    v8f s[4];
#pragma unroll
    for (int t = 0; t < 4; ++t) {
      v8f acc = {};
#pragma unroll
      for (int kk = 0; kk < 16; ++kk) {
        const int d = kk * 4 + khalf * 2;
        v2f bf;                          // B(K=d, N=key) = K[key][d]
        bf.x = Ks[(t * 16 + lnlo) * ALDV + d];
        bf.y = Ks[(t * 16 + lnlo) * ALDV + d + 1];
        acc = wmma4(aq[kk], bf, acc);
      }
      s[t] = acc;
    }

    // ---- scale (1/sqrt(64)) + causal mask (diagonal tile only)
    const float sc = 0.125f;
    if (kt == qt) {
#pragma unroll
      for (int t = 0; t < 4; ++t) {
        const int kcol = kbase + t * 16 + lnlo;
#pragma unroll
        for (int r = 0; r < 8; ++r) {
          const int qrow = qbase + r + khalf * 8;
          s[t][r] = (kcol > qrow) ? -__builtin_inff() : s[t][r] * sc;
        }
      }
    } else {
#pragma unroll
      for (int t = 0; t < 4; ++t)
#pragma unroll
        for (int r = 0; r < 8; ++r) s[t][r] *= sc;
    }

    // ---- online softmax (rows live across lanes 0-15 / 16-31 in C layout)
    float alpha[8];
#pragma unroll
    for (int r = 0; r < 8; ++r) {
      float v = fmaxf(fmaxf(s[0][r], s[1][r]), fmaxf(s[2][r], s[3][r]));
      v = fmaxf(v, __shfl_xor(v, 1, 32));
      v = fmaxf(v, __shfl_xor(v, 2, 32));
      v = fmaxf(v, __shfl_xor(v, 4, 32));
      v = fmaxf(v, __shfl_xor(v, 8, 32));
      const float mn = fmaxf(mrow[r], v);
      alpha[r] = expf(mrow[r] - mn);     // exp(-inf)=0 on first tile
      mrow[r]  = mn;
    }
    float rs[8];
#pragma unroll
    for (int r = 0; r < 8; ++r) rs[r] = 0.f;
#pragma unroll
    for (int t = 0; t < 4; ++t)
#pragma unroll
      for (int r = 0; r < 8; ++r) {
        const float p = expf(s[t][r] - mrow[r]);   // masked -> exp(-inf)=0
        s[t][r] = p;
        rs[r] += p;
      }
#pragma unroll
    for (int r = 0; r < 8; ++r) {
      float v = rs[r];
      v += __shfl_xor(v, 1, 32);
      v += __shfl_xor(v, 2, 32);
      v += __shfl_xor(v, 4, 32);
      v += __shfl_xor(v, 8, 32);
      lrow[r] = lrow[r] * alpha[r] + v;
      o0[r] *= alpha[r]; o1[r] *= alpha[r];
      o2[r] *= alpha[r]; o3[r] *= alpha[r];
    }

    // ---- C-layout -> A-layout via wave-private LDS strip (in-order DS ops)
#pragma unroll
    for (int t = 0; t < 4; ++t)
#pragma unroll
      for (int r = 0; r < 8; ++r)
        Pw[(r + khalf * 8) * ALDK + t * 16 + lnlo] = s[t][r];

    // ---- O += P @ V : 16x64 += (16x64)(64x64), 64 WMMAs
#pragma unroll
    for (int kk = 0; kk < 16; ++kk) {
      const int kd = kk * 4 + khalf * 2;
      v2f ap;
      ap.x = Pw[lnlo * ALDK + kd];
      ap.y = Pw[lnlo * ALDK + kd + 1];
      v2f b0, b1, b2, b3;
      b0.x = Vs[(kd + 0) * ALDV + 0  + lnlo]; b0.y = Vs[(kd + 1) * ALDV + 0  + lnlo];
      b1.x = Vs[(kd + 0) * ALDV + 16 + lnlo]; b1.y = Vs[(kd + 1) * ALDV + 16 + lnlo];
      b2.x = Vs[(kd + 0) * ALDV + 32 + lnlo]; b2.y = Vs[(kd + 1) * ALDV + 32 + lnlo];
      b3.x = Vs[(kd + 0) * ALDV + 48 + lnlo]; b3.y = Vs[(kd + 1) * ALDV + 48 + lnlo];
      o0 = wmma4(ap, b0, o0);
      o1 = wmma4(ap, b1, o1);
      o2 = wmma4(ap, b2, o2);
      o3 = wmma4(ap, b3, o3);
    }
  }

  // ---- normalize and write ctx (overwrites our own Q slice only: safe alias)
#pragma unroll
  for (int r = 0; r < 8; ++r) {
    const float inv = 1.0f / lrow[r];
    const size_t ro = bo + (size_t)(qbase + r + khalf * 8) * kD + hc;
    Ctx[ro + 0  + lnlo] = o0[r] * inv;
    Ctx[ro + 16 + lnlo] = o1[r] * inv;
    Ctx[ro + 32 + lnlo] = o2[r] * inv;
    Ctx[ro + 48 + lnlo] = o3[r] * inv;
  }
}

// ---------------------------------------------------------------------------
extern "C" void kernel_launch(void* const* d_in, const int* in_sizes, int n_in,
                              void* d_out, int out_size, void* d_ws, size_t ws_size,
                              hipStream_t stream) {
  const float* x   = (const float*)d_in[0];
  const float* Wq  = (const float*)d_in[1];
  const float* Wk  = (const float*)d_in[2];
  const float* Wv  = (const float*)d_in[3];
  const float* Wo  = (const float*)d_in[4];
  const float* bo_ = (const float*)d_in[5];
  float* out = (float*)d_out;

  const size_t qkvElems = (size_t)kB * kS * kD;
  float* Q = (float*)d_ws;
  float* K = Q + qkvElems;
  float* V = K + qkvElems;
  // ctx gets its own region if the workspace allows, else aliases Q (safe:
  // each attention WG reads its Q slice before writing the same ctx slice).
  float* Ctx = (ws_size >= 4 * qkvElems * sizeof(float)) ? (V + qkvElems) : Q;

  const int M = kB * kS;                        // 8192
  dim3 gGemm(kD / GTN, M / GTM);                // 12 x 128
  gemm_bias_kernel<<<gGemm, 256, 0, stream>>>(x, Wq, nullptr, Q, M, kD, kD);
  gemm_bias_kernel<<<gGemm, 256, 0, stream>>>(x, Wk, nullptr, K, M, kD, kD);
  gemm_bias_kernel<<<gGemm, 256, 0, stream>>>(x, Wv, nullptr, V, M, kD, kD);

  dim3 gAttn(kS / 64, kH, kB);                  // 32 x 12 x 4
  flash_attn_kernel<<<gAttn, 128, 0, stream>>>(Q, K, V, Ctx);

  gemm_bias_kernel<<<gGemm, 256, 0, stream>>>(Ctx, Wo, bo_, out, M, kD, kD);
}